// ChannelAttention_SNN_36670430773409
// MI455X (gfx1250) — compile-verified
//
#include <hip/hip_runtime.h>

typedef __attribute__((ext_vector_type(2))) float v2f;
typedef __attribute__((ext_vector_type(4))) float v4f;
typedef __attribute__((ext_vector_type(8))) float v8f;

namespace {
constexpr int T  = 4;
constexpr int B  = 16;
constexpr int C  = 512;
constexpr int N  = 1024;
constexpr int CR = 64;
constexpr int TB = 64;                          // T*B rows after GAP
constexpr size_t STRIDE_T = (size_t)B * C * N;  // floats between time slices
constexpr float BN_EPS = 1e-5f;
}

// ---------------------------------------------------------------------------
// Kernel 1: fused LIF multistep + global average pool over N.
// One block per (b,c) row (8192 blocks x 256 threads). Each thread owns 4
// contiguous elements (128-bit NT loads), carries the membrane potential
// across the 4 timesteps, counts spikes per t, then wave32-shuffle + LDS
// reduce to the mean. This kernel carries ~100% of the memory traffic
// (128 MB, fully coalesced, single-use -> non-temporal).
// ---------------------------------------------------------------------------
__global__ void lif_gap_kernel(const float* __restrict__ x, float* __restrict__ g) {
    const int bc   = blockIdx.x;           // b*C + c
    const int tid  = threadIdx.x;
    const int lane = tid & 31;
    const int wid  = tid >> 5;

    const v4f* p = reinterpret_cast<const v4f*>(x + (size_t)bc * N) + tid;

    float v[4] = {0.f, 0.f, 0.f, 0.f};
    float cnt[T];
#pragma unroll
    for (int t = 0; t < T; ++t) {
        v4f xv = __builtin_nontemporal_load(p + (size_t)t * (STRIDE_T / 4));
        float ct = 0.f;
#pragma unroll
        for (int j = 0; j < 4; ++j) {
            v[j] = 0.5f * (v[j] + xv[j]);              // v += (x - v)/tau, tau=2
            float s = (v[j] >= 1.0f) ? 1.0f : 0.0f;    // Heaviside(v - v_th)
            ct  += s;
            v[j] = (1.0f - s) * v[j];                  // hard reset to 0
        }
        cnt[t] = ct;
    }

    // wave32 shuffle reduction, then cross-wave via LDS
#pragma unroll
    for (int t = 0; t < T; ++t)
#pragma unroll
        for (int off = 16; off > 0; off >>= 1)
            cnt[t] += __shfl_down(cnt[t], off, 32);

    __shared__ float red[8][T];
    if (lane == 0) {
#pragma unroll
        for (int t = 0; t < T; ++t) red[wid][t] = cnt[t];
    }
    __syncthreads();

    if (tid < T) {
        float a = 0.f;
#pragma unroll
        for (int w = 0; w < 8; ++w) a += red[w][tid];
        const int b = bc >> 9;          // bc / C
        const int c = bc & (C - 1);     // bc % C
        g[((size_t)(tid * B + b) * C) + c] = a * (1.0f / N);
    }
}

// ---------------------------------------------------------------------------
// Kernel 2/4: fp32 WMMA GEMM  D[M x Nn] = A[M x K] * Bt[Nn x K]^T + bias.
// One wave per 16x16 output tile; K consumed 4 at a time via
// V_WMMA_F32_16X16X4_F32.  The K loop trip count is a *scalar* (uniform) so
// no exec masking is generated around the WMMAs (ISA 7.12: EXEC must be all
// ones).  Operand layouts per CDNA5 ISA 7.12.2:
//   A 16x4 : lanes 0-15 -> K={0,1} in v0/v1, lanes 16-31 -> K={2,3}
//   B 4x16 : N = lane&15 across lanes, same K split across lane halves
//   C/D    : row M = vgpr + 8*(lane>=16), col N = lane&15
// ---------------------------------------------------------------------------
__global__ void wmma_gemm_kernel(const float* __restrict__ A,
                                 const float* __restrict__ Bt,
                                 const float* __restrict__ bias,
                                 float* __restrict__ D,
                                 int K4 /* = K/4, uniform */, int K,
                                 int ldd, int ntn) {
    const int lane = threadIdx.x;          // 32 threads = 1 wave
    const int mi   = blockIdx.x / ntn;
    const int ni   = blockIdx.x % ntn;
    const int l    = lane & 15;
    const int half = lane >> 4;

    const int m = mi * 16 + l;
    const int n = ni * 16 + l;
    // per-lane base: this lane's K pair starts at half*2
    const float* arow = A  + (size_t)m * K + half * 2;
    const float* brow = Bt + (size_t)n * K + half * 2;

    v8f acc = {};
#pragma unroll 4
    for (int kk = 0; kk < K4; ++kk) {
        v2f a = *reinterpret_cast<const v2f*>(arow + 4 * kk);
        v2f b = *reinterpret_cast<const v2f*>(brow + 4 * kk);
        acc = __builtin_amdgcn_wmma_f32_16x16x4_f32(
            /*neg_a=*/false, a, /*neg_b=*/false, b,
            /*c_mod=*/(short)0, acc, /*reuse_a=*/false, /*reuse_b=*/false);
    }

    const float bv = bias[n];
#pragma unroll
    for (int r = 0; r < 8; ++r) {
        const int mr = mi * 16 + r + half * 8;
        D[(size_t)mr * ldd + n] = acc[r] + bv;
    }
}

// ---------------------------------------------------------------------------
// Kernel 3/5: training-mode BatchNorm over the TB=64 batch rows, one thread
// per column (population variance, matches jnp.var ddof=0).  In-place safe.
// ---------------------------------------------------------------------------
__global__ void bn_kernel(const float* __restrict__ h,
                          const float* __restrict__ gamma,
                          const float* __restrict__ beta,
                          float* __restrict__ out, int cols) {
    const int j = blockIdx.x * blockDim.x + threadIdx.x;
    if (j >= cols) return;
    float s = 0.f, sq = 0.f;
    for (int m = 0; m < TB; ++m) {
        float xv = h[(size_t)m * cols + j];
        s  += xv;
        sq += xv * xv;
    }
    const float mu  = s  * (1.0f / TB);
    const float var = sq * (1.0f / TB) - mu * mu;
    const float rs  = rsqrtf(var + BN_EPS);
    const float ga  = gamma[j] * rs;
    const float be  = beta[j] - mu * ga;
    for (int m = 0; m < TB; ++m)
        out[(size_t)m * cols + j] = h[(size_t)m * cols + j] * ga + be;
}

// ---------------------------------------------------------------------------
extern "C" void kernel_launch(void* const* d_in, const int* in_sizes, int n_in,
                              void* d_out, int out_size, void* d_ws, size_t ws_size,
                              hipStream_t stream) {
    const float* x      = (const float*)d_in[0];
    const float* w1     = (const float*)d_in[1];   // [CR, C]
    const float* b1     = (const float*)d_in[2];
    const float* gamma1 = (const float*)d_in[3];
    const float* beta1  = (const float*)d_in[4];
    const float* w2     = (const float*)d_in[5];   // [C, CR]
    const float* b2     = (const float*)d_in[6];
    const float* gamma2 = (const float*)d_in[7];
    const float* beta2  = (const float*)d_in[8];
    float* out = (float*)d_out;                    // [TB, C]

    float* g  = (float*)d_ws;                      // [TB, C]   128 KB
    float* h1 = g  + (size_t)TB * C;               // [TB, CR]   16 KB
    float* h2 = h1 + (size_t)TB * CR;              // [TB, C]   128 KB

    // 1) LIF + GAP (bandwidth-dominant streaming pass)
    lif_gap_kernel<<<B * C, 256, 0, stream>>>(x, g);

    // 2) h1 = g @ w1^T + b1   (M=64, N=64, K=512) -> 16 tiles
    wmma_gemm_kernel<<<(TB / 16) * (CR / 16), 32, 0, stream>>>(g, w1, b1, h1,
                                                               C / 4, C, CR, CR / 16);
    // 3) BN1 (in place)
    bn_kernel<<<1, CR, 0, stream>>>(h1, gamma1, beta1, h1, CR);

    // 4) h2 = h1 @ w2^T + b2  (M=64, N=512, K=64) -> 128 tiles
    wmma_gemm_kernel<<<(TB / 16) * (C / 16), 32, 0, stream>>>(h1, w2, b2, h2,
                                                              CR / 4, CR, C, C / 16);
    // 5) BN2 -> output
    bn_kernel<<<2, 256, 0, stream>>>(h2, gamma2, beta2, out, C);
}